// _SimpleNeighborhoodPooling_65781719106309
// MI455X (gfx1250) — compile-verified
//
#include <hip/hip_runtime.h>

// Problem constants (from reference setup_inputs)
#define Bc 4
#define Nc 65536
#define Sc 4096
#define Kc 32
#define Dc 256
#define LN_EPS 1e-5f

typedef __attribute__((ext_vector_type(2))) float v2f;
typedef __attribute__((ext_vector_type(8))) float v8f;

struct alignas(16) F4 { float x, y, z, w; };

__device__ __forceinline__ F4 ld4(const float* p) {
    return *reinterpret_cast<const F4*>(p);
}

__device__ __forceinline__ float wave_red32(float v) {
#pragma unroll
    for (int off = 16; off >= 1; off >>= 1)
        v += __shfl_xor(v, off, 32);
    return v;
}

// ---------------------------------------------------------------------------
// Kernel 1: FiLM projection  gb[4,512] = task_emb[4,256] @ film_w[256,512] + film_b
// Exact f32 GEMM via V_WMMA_F32_16X16X4_F32. One block, 32 waves, one 16-wide
// N-tile per wave, K looped 4 at a time.
// A layout (16x4 f32): lane&15 = M; VGPR0 = K{0 | 2}, VGPR1 = K{1 | 3} by lane half.
// B layout (4x16 f32): lane&15 = N; same K split by lane half.
// C layout: VGPR r holds M = r + 8*(lane>=16), N = lane&15.
// ---------------------------------------------------------------------------
__global__ __launch_bounds__(1024) void film_wmma_kernel(
    const float* __restrict__ task_emb,   // [4,256]
    const float* __restrict__ film_w,     // [256,512]
    const float* __restrict__ film_b,     // [512]
    float* __restrict__ gb)               // [4,512] out (workspace)
{
    const int lane   = threadIdx.x & 31;
    const int wave   = threadIdx.x >> 5;      // 0..31
    const int n_base = wave * 16;             // 32 tiles cover 512 cols
    const int m      = lane & 15;
    const int kh     = lane >> 4;             // 0: K+0/K+1, 1: K+2/K+3
    const int n      = n_base + (lane & 15);
    const float mval = (m < 4) ? 1.0f : 0.0f; // zero padded M rows (no divergence)
    const int  mrow  = (m & 3) * Dc;

    v8f c = {};
    for (int kc = 0; kc < Dc; kc += 4) {
        const int k0 = kc + 2 * kh;
        v2f a, b;
        a.x = task_emb[mrow + k0]     * mval;
        a.y = task_emb[mrow + k0 + 1] * mval;
        b.x = film_w[(size_t)k0       * 512 + n];
        b.y = film_w[(size_t)(k0 + 1) * 512 + n];
        c = __builtin_amdgcn_wmma_f32_16x16x4_f32(
                /*neg_a=*/false, a, /*neg_b=*/false, b,
                /*c_mod=*/(short)0, c, /*reuse_a=*/false, /*reuse_b=*/false);
    }

    if (lane < 16) {            // lanes 0..15 hold M = 0..7 in VGPRs 0..7
        const float fb = film_b[n];
#pragma unroll
        for (int r = 0; r < 4; ++r)            // batches 0..3
            gb[r * 512 + n] = c[r] + fb;
    }
}

// ---------------------------------------------------------------------------
// Kernel 2: fused gather + masked mean + FiLM + residual + LayerNorm.
// One wave (32 lanes) per supernode; lane k owns neighbor k's idx/mask.
// Mask bits are wave-uniform -> branchy skip is free and halves gather bytes.
// Each active k: whole wave reads the full 1KB row as coalesced b128 loads.
// ---------------------------------------------------------------------------
__global__ __launch_bounds__(256) void pool_film_ln_kernel(
    const float* __restrict__ point_feat,       // [4,65536,256]
    const long long* __restrict__ nidx,         // [4,4096,32] int64
    const unsigned char* __restrict__ nmask,    // [4,4096,32] bool
    const float* __restrict__ sinit,            // [4,4096,256]
    const float* __restrict__ gb,               // [4,512] gamma|beta
    const float* __restrict__ ln_w,             // [256]
    const float* __restrict__ ln_b,             // [256]
    float* __restrict__ out)                    // [4,4096,256]
{
    const int lane = threadIdx.x & 31;
    const int wave = threadIdx.x >> 5;                  // 0..7
    const int g    = blockIdx.x * 8 + wave;             // global supernode id
    const int b    = g >> 12;                           // / 4096
    const float* __restrict__ pf = point_feat + (size_t)b * Nc * Dc;

    // lane k holds neighbor k's index + mask
    const int           my_idx = (int)nidx[(size_t)g * Kc + lane];
    const unsigned char my_m   = nmask[(size_t)g * Kc + lane];
    const unsigned      mb     = (unsigned)__ballot(my_m != 0);
    const int           cnt    = __popc(mb);

    // warm L2/L0 with active row starts (global_prefetch_b8)
    if (my_m) {
        const float* rp = pf + (size_t)my_idx * Dc;
        __builtin_prefetch(rp, 0, 0);
        __builtin_prefetch(rp + 128, 0, 0);
    }

    // masked sum over K: each lane covers d = 4*lane..4*lane+3 and +128
    F4 acc0 = {0.f, 0.f, 0.f, 0.f}, acc1 = {0.f, 0.f, 0.f, 0.f};
    for (int k = 0; k < Kc; ++k) {
        if ((mb >> k) & 1u) {                          // wave-uniform branch
            const int row = __shfl(my_idx, k, 32);
            const float* rp = pf + (size_t)row * Dc;
            const F4 v0 = ld4(rp + lane * 4);
            const F4 v1 = ld4(rp + 128 + lane * 4);
            acc0.x += v0.x; acc0.y += v0.y; acc0.z += v0.z; acc0.w += v0.w;
            acc1.x += v1.x; acc1.y += v1.y; acc1.z += v1.z; acc1.w += v1.w;
        }
    }
    const float inv = 1.0f / (float)(cnt > 0 ? cnt : 1);

    // FiLM:  agg*(1+gamma) + beta ; then residual with supernode_init
    const float* gbb = gb + b * 512;
    const F4 ga0 = ld4(gbb + lane * 4);
    const F4 ga1 = ld4(gbb + 128 + lane * 4);
    const F4 be0 = ld4(gbb + 256 + lane * 4);
    const F4 be1 = ld4(gbb + 384 + lane * 4);
    const float* sip = sinit + (size_t)g * Dc;
    const F4 si0 = ld4(sip + lane * 4);
    const F4 si1 = ld4(sip + 128 + lane * 4);

    F4 x0, x1;
    x0.x = si0.x + (acc0.x * inv) * (1.0f + ga0.x) + be0.x;
    x0.y = si0.y + (acc0.y * inv) * (1.0f + ga0.y) + be0.y;
    x0.z = si0.z + (acc0.z * inv) * (1.0f + ga0.z) + be0.z;
    x0.w = si0.w + (acc0.w * inv) * (1.0f + ga0.w) + be0.w;
    x1.x = si1.x + (acc1.x * inv) * (1.0f + ga1.x) + be1.x;
    x1.y = si1.y + (acc1.y * inv) * (1.0f + ga1.y) + be1.y;
    x1.z = si1.z + (acc1.z * inv) * (1.0f + ga1.z) + be1.z;
    x1.w = si1.w + (acc1.w * inv) * (1.0f + ga1.w) + be1.w;

    // LayerNorm over D=256 (8 values/lane, wave32 tree reduction)
    float lsum = x0.x + x0.y + x0.z + x0.w + x1.x + x1.y + x1.z + x1.w;
    const float mu = wave_red32(lsum) * (1.0f / 256.0f);

    const float d0x = x0.x - mu, d0y = x0.y - mu, d0z = x0.z - mu, d0w = x0.w - mu;
    const float d1x = x1.x - mu, d1y = x1.y - mu, d1z = x1.z - mu, d1w = x1.w - mu;
    float lsq = d0x*d0x + d0y*d0y + d0z*d0z + d0w*d0w
              + d1x*d1x + d1y*d1y + d1z*d1z + d1w*d1w;
    const float var = wave_red32(lsq) * (1.0f / 256.0f);
    const float rs  = 1.0f / sqrtf(var + LN_EPS);

    const F4 w0 = ld4(ln_w + lane * 4);
    const F4 w1 = ld4(ln_w + 128 + lane * 4);
    const F4 lb0 = ld4(ln_b + lane * 4);
    const F4 lb1 = ld4(ln_b + 128 + lane * 4);

    F4 y0, y1;
    y0.x = d0x * rs * w0.x + lb0.x;
    y0.y = d0y * rs * w0.y + lb0.y;
    y0.z = d0z * rs * w0.z + lb0.z;
    y0.w = d0w * rs * w0.w + lb0.w;
    y1.x = d1x * rs * w1.x + lb1.x;
    y1.y = d1y * rs * w1.y + lb1.y;
    y1.z = d1z * rs * w1.z + lb1.z;
    y1.w = d1w * rs * w1.w + lb1.w;

    float* op = out + (size_t)g * Dc;
    *reinterpret_cast<F4*>(op + lane * 4)       = y0;
    *reinterpret_cast<F4*>(op + 128 + lane * 4) = y1;
}

// ---------------------------------------------------------------------------
extern "C" void kernel_launch(void* const* d_in, const int* in_sizes, int n_in,
                              void* d_out, int out_size, void* d_ws, size_t ws_size,
                              hipStream_t stream) {
    const float*         point_feat = (const float*)d_in[0];
    const long long*     nidx       = (const long long*)d_in[1];
    const unsigned char* nmask      = (const unsigned char*)d_in[2];
    const float*         sinit      = (const float*)d_in[3];
    const float*         task_emb   = (const float*)d_in[4];
    const float*         film_w     = (const float*)d_in[5];
    const float*         film_b     = (const float*)d_in[6];
    const float*         ln_w       = (const float*)d_in[7];
    const float*         ln_b       = (const float*)d_in[8];
    float* out = (float*)d_out;
    float* gb  = (float*)d_ws;          // [4,512] gamma|beta scratch (8 KB)

    film_wmma_kernel<<<1, 1024, 0, stream>>>(task_emb, film_w, film_b, gb);

    const int num_super = Bc * Sc;      // 16384 supernodes, 8 per block
    pool_film_ln_kernel<<<num_super / 8, 256, 0, stream>>>(
        point_feat, nidx, nmask, sinit, gb, ln_w, ln_b, out);
}